// PolyaTree_72103910965826
// MI455X (gfx1250) — compile-verified
//
#include <hip/hip_runtime.h>
#include <hip/hip_bf16.h>
#include <math.h>

#define L_LVL   10
#define DIM     64
#define M_NODES 1023
#define N_INT   511      // interior nodes 0..510
#define N_LEAF  512      // leaves 511..1022
#define EPS_SPLIT 1e-07f
#define CLAMP_V   1e-05f

typedef unsigned int u32;
typedef unsigned long long u64;
typedef u32 v4u __attribute__((ext_vector_type(4)));
typedef int  v4i __attribute__((ext_vector_type(4)));
typedef int  v8i __attribute__((ext_vector_type(8)));

// ---------------------------------------------------------------------------
// Kernel 1: build per-node split table and per-leaf term table.
// One thread per (node, dim): walk the heap path given by bits of (node+1),
// maintaining [lo, up] and the cumulative product of samples along the path.
//   interior node p : bnd[p*64+d]          = lo + beta_p * (up - lo)
//   leaf node     q : term[(q-511)*64+d]   = log(max(cumprod,1e-5) - log(max(up-lo,1e-5)))
// Full-precision logf: this runs once over 65K threads, accuracy > speed here.
// ---------------------------------------------------------------------------
__global__ void __launch_bounds__(256)
polya_setup(const float* __restrict__ samples,  // (DIM, M) : samples[d*M + m]
            float* __restrict__ bnd,            // (511, 64)
            float* __restrict__ term)           // (512, 64)
{
    int idx = blockIdx.x * blockDim.x + threadIdx.x;
    if (idx >= M_NODES * DIM) return;
    int node = idx >> 6;     // 0..1022
    int d    = idx & 63;

    const float* sd = samples + d * M_NODES;

    int   np1   = node + 1;
    int   depth = 31 - __clz(np1);   // level of node (root = 0)
    float lo = 0.0f, up = 1.0f;
    int   cur = 0;
    float cp  = sd[0];               // product includes the root sample

    for (int k = depth - 1; k >= 0; --k) {
        int   b     = (np1 >> k) & 1;          // 0 = go left, 1 = go right
        float beta  = sd[cur];
        float split = lo + beta * (up - lo);
        if (b == 0) { up = split;             cur = 2 * cur + 1; }
        else        { lo = split + EPS_SPLIT; cur = 2 * cur + 2; }
        cp *= sd[cur];
    }
    // cur == node here, with its interval [lo, up] and path product cp.
    if (node < N_INT) {
        float beta = sd[node];
        bnd[node * DIM + d] = lo + beta * (up - lo);
    } else {
        float width = up - lo;
        float logB  = logf(fmaxf(width, CLAMP_V));
        term[(node - N_INT) * DIM + d] = logf(fmaxf(cp, CLAMP_V) - logB);
    }
}

// ---------------------------------------------------------------------------
// Kernel 2: per-sample tree descent.
//   - bnd table (511*64 floats = 127.75 KB) staged into LDS via the Tensor
//     Data Mover (tensor_load_to_lds + s_wait_tensorcnt 0).
//   - wave32 mapping: lane L owns dims 2L and 2L+1 (two chains/lane),
//     4 samples in flight per wave => 8 independent 9-deep dependent
//     LDS-gather chains for latency hiding; x loads are one b64/lane
//     (256 B per wave instruction, fully coalesced).
//   - LDS layout bnd[node*64 + d]: bank = d; chain A banks = even lanes'
//     dims (all distinct), chain B = odd => conflict-free gathers.
//   - reduction over 64 dims: chainA+chainB then 5x __shfl_xor; one
//     coalesced b32 store with lanes 0..3 active writes 4 outputs.
// ---------------------------------------------------------------------------
__global__ void __launch_bounds__(256)
polya_main(const float* __restrict__ x,     // (N, 64)
           const float* __restrict__ bnd,   // (511, 64) in global (TDM source)
           const float* __restrict__ term,  // (512, 64) in global (L2-hot)
           float* __restrict__ out,         // (N,)
           int N)
{
    extern __shared__ float s_bnd[];   // 511*64 floats = 130816 bytes

    // ---- TDM: DMA the boundary table global -> LDS (one descriptor) ----
    if (threadIdx.x == 0) {
        const u32 NELEM = N_INT * DIM;              // 32704 elements (f32)
        u32 lds_addr = (u32)(size_t)&s_bnd[0];      // low 32 bits = LDS byte offset
        u64 ga       = (u64)(const void*)bnd;

        v4u g0;
        g0.x = 1u;                                   // count=1, user mode, no gather
        g0.y = lds_addr;                             // lds_addr[31:0]
        g0.z = (u32)(ga & 0xFFFFFFFFull);            // global_addr[31:0]
        g0.w = (u32)((ga >> 32) & 0x1FFFFFFull)      // global_addr[56:32]
             | (2u << 30);                           // type = 2 ("image")

        v8i g1;
        g1[0] = (int)(2u << 16);                     // data_size = 2 -> 4 bytes; mask=0
        g1[1] = (int)((NELEM & 0xFFFFu) << 16);      // tensor_dim0[15:0]
        g1[2] = (int)(((NELEM >> 16) & 0xFFFFu)      // tensor_dim0[31:16]
             | (1u << 16));                          // tensor_dim1 = 1 (low 16)
        g1[3] = (int)((NELEM & 0xFFFFu) << 16);      // tile_dim0 = 32704
        g1[4] = 1;                                   // tile_dim1 = 1, tile_dim2 = 0
        g1[5] = (int)NELEM;                          // tensor_dim0_stride[31:0]
        g1[6] = 0;                                   // stride[47:32]=0, dim1_stride lo=0
        g1[7] = 0;

        v4i g2 = {0, 0, 0, 0};                       // <=2-D tensor: groups 2/3 unused
        v4i g3 = {0, 0, 0, 0};
        v8i g4 = {0, 0, 0, 0, 0, 0, 0, 0};           // extra group (clang-23 6-arg form)
        __builtin_amdgcn_tensor_load_to_lds(g0, g1, g2, g3, g4, 0);
    }
    __builtin_amdgcn_s_wait_tensorcnt(0);            // no-op for waves that didn't issue
    __syncthreads();                                  // LDS table visible to all waves

    const int lane  = threadIdx.x & 31;
    const int wave  = threadIdx.x >> 5;
    const int wpb   = blockDim.x >> 5;
    const int gwave = blockIdx.x * wpb + wave;
    const int tot   = gridDim.x * wpb;

    const int dA = 2 * lane;          // even dim owned by this lane
    const int dB = 2 * lane + 1;      // odd dim owned by this lane

    for (int n0 = gwave * 4; n0 < N; n0 += tot * 4) {
        float xA[4], xB[4];
        int   curA[4], curB[4];
#pragma unroll
        for (int s = 0; s < 4; ++s) {
            int n = n0 + s;                             // N multiple of 4 -> in range
            const float2 xv =
                *(const float2*)(x + (size_t)n * DIM + dA);  // one b64, coalesced
            xA[s] = xv.x;
            xB[s] = xv.y;
            curA[s] = 0;
            curB[s] = 0;
        }
        // 9-level descent; 8 independent dependent-load chains.
#pragma unroll
        for (int lvl = 0; lvl < L_LVL - 1; ++lvl) {
#pragma unroll
            for (int s = 0; s < 4; ++s) {
                float spA = s_bnd[curA[s] * DIM + dA];   // bank = 2*lane   (distinct)
                float spB = s_bnd[curB[s] * DIM + dB];   // bank = 2*lane+1 (distinct)
                curA[s] = (xA[s] <= spA) ? (2 * curA[s] + 1) : (2 * curA[s] + 2);
                curB[s] = (xB[s] <= spB) ? (2 * curB[s] + 1) : (2 * curB[s] + 2);
            }
        }
        // Leaf lookup + reduction over 64 dims (result lands in every lane).
        float vs[4];
#pragma unroll
        for (int s = 0; s < 4; ++s) {
            float v = term[(curA[s] - N_INT) * DIM + dA]
                    + term[(curB[s] - N_INT) * DIM + dB];
#pragma unroll
            for (int off = 16; off > 0; off >>= 1)
                v += __shfl_xor(v, off, 32);
            vs[s] = v * (1.0f / 64.0f);
        }
        // One coalesced 16B store: lanes 0..3 write out[n0+lane].
        float vsel = (lane == 1) ? vs[1] : (lane == 2) ? vs[2] : (lane == 3) ? vs[3] : vs[0];
        if (lane < 4) out[n0 + lane] = vsel;
    }
}

// ---------------------------------------------------------------------------
// Host-side launcher (graph-capture safe: only kernel launches on `stream`).
//   d_in[0] = x        (N*64 f32)
//   d_in[1] = samples  (64*1023 f32)
//   d_out   = out      (N f32)
//   d_ws    : [0, 130816)        bnd  (511*64 f32)
//             [131072, 262144)   term (512*64 f32)
// ---------------------------------------------------------------------------
extern "C" void kernel_launch(void* const* d_in, const int* in_sizes, int n_in,
                              void* d_out, int out_size, void* d_ws, size_t ws_size,
                              hipStream_t stream)
{
    const float* x       = (const float*)d_in[0];
    const float* samples = (const float*)d_in[1];
    float*       out     = (float*)d_out;

    float* bnd  = (float*)d_ws;                       // 511*64 floats
    float* term = (float*)d_ws + (N_INT * DIM + DIM); // 32768 floats in -> 128 KB offset

    int N = in_sizes[0] / DIM;                        // 131072

    // Kernel 1: 1023*64 = 65472 (node, dim) threads.
    {
        int threads = 256;
        int blocks  = (M_NODES * DIM + threads - 1) / threads;
        polya_setup<<<blocks, threads, 0, stream>>>(samples, bnd, term);
    }

    // Kernel 2: 1024 blocks x 256 threads (8 waves), 128 KB dynamic LDS each
    // (2 blocks per 320 KB WGP -> 16 waves/WGP). 8192 waves x 4 samples/iter.
    {
        size_t shmem = (size_t)(N_INT * DIM) * sizeof(float);   // 130816 B
        polya_main<<<1024, 256, shmem, stream>>>(x, bnd, term, out, N);
    }
}